// GCN_64948495450765
// MI455X (gfx1250) — compile-verified
//
#include <hip/hip_runtime.h>
#include <hip/hip_bf16.h>

typedef float v2f __attribute__((ext_vector_type(2)));
typedef float v8f __attribute__((ext_vector_type(8)));

#define N_NODES   16384
#define N_EDGES   524288
#define D_IN      512
#define D_HID     256
#define D_OUT2    64

// ---------------------------------------------------------------------------
// Zero-fill helper
// ---------------------------------------------------------------------------
__global__ void fill_zero_kernel(float* __restrict__ p, size_t n) {
    size_t i = (size_t)blockIdx.x * blockDim.x + threadIdx.x;
    if (i < n) p[i] = 0.0f;
}

// ---------------------------------------------------------------------------
// GEMM 1: h0[16384,256] = x[16384,512] @ W1[512,256]   (f32 WMMA 16x16x4)
// One wave32 per 16x16 output tile. K loop = 512/4 = 128 WMMA ops.
// ---------------------------------------------------------------------------
__global__ void gemm_xW1_kernel(const float* __restrict__ X,
                                const float* __restrict__ W,
                                float* __restrict__ H0) {
    const int wave  = (int)((blockIdx.x * blockDim.x + threadIdx.x) >> 5);
    const int lane  = threadIdx.x & 31;
    const int tileM = wave >> 4;          // 256/16 = 16 N-tiles
    const int tileN = wave & 15;
    const int ml    = lane & 15;
    const int half  = lane >> 4;          // 0 or 1
    const int kh    = half << 1;          // K sub-offset within 4
    const size_t m  = (size_t)(tileM << 4) + ml;   // A row
    const int    n  = (tileN << 4) + ml;           // B/D col

    v8f c = {};
    for (int k0 = 0; k0 < D_IN; k0 += 4) {
        // A: 16x4 tile, lane holds X[m][k0+kh .. k0+kh+1] (contiguous -> b64)
        v2f a = *(const v2f*)(X + m * D_IN + k0 + kh);
        // B: 4x16 tile, lane holds W[k0+kh][n], W[k0+kh+1][n]
        v2f b;
        b.x = W[(size_t)(k0 + kh)     * D_HID + n];
        b.y = W[(size_t)(k0 + kh + 1) * D_HID + n];
        c = __builtin_amdgcn_wmma_f32_16x16x4_f32(false, a, false, b,
                                                  (short)0, c, false, false);
    }
    // D: VGPR v -> row = tileM*16 + v + 8*half, col = n
    const int rbase = (tileM << 4) + (half << 3);
#pragma unroll
    for (int v = 0; v < 8; ++v)
        H0[(size_t)(rbase + v) * D_HID + n] = c[v];
}

// ---------------------------------------------------------------------------
// SpMM 1: h1[row] += val * h0[col], 256 channels. One block per edge.
// ---------------------------------------------------------------------------
__global__ void spmm256_kernel(const int*   __restrict__ erow,
                               const int*   __restrict__ ecol,
                               const float* __restrict__ eval,
                               const float* __restrict__ H0,
                               float*       __restrict__ H1) {
    const int e = blockIdx.x;
    const int c = threadIdx.x;            // 256 channels
    const int r = erow[e];
    const int s = ecol[e];
    const float v = eval[e];
    atomicAdd(&H1[(size_t)r * D_HID + c], v * H0[(size_t)s * D_HID + c]);
}

// ---------------------------------------------------------------------------
// GEMM 2: z0[16384,64] = relu(h1)[16384,256] @ W2[256,64]   (ReLU fused on A)
// ---------------------------------------------------------------------------
__global__ void gemm_hW2_kernel(const float* __restrict__ H1,
                                const float* __restrict__ W,
                                float* __restrict__ Z0) {
    const int wave  = (int)((blockIdx.x * blockDim.x + threadIdx.x) >> 5);
    const int lane  = threadIdx.x & 31;
    const int tileM = wave >> 2;          // 64/16 = 4 N-tiles
    const int tileN = wave & 3;
    const int ml    = lane & 15;
    const int half  = lane >> 4;
    const int kh    = half << 1;
    const size_t m  = (size_t)(tileM << 4) + ml;
    const int    n  = (tileN << 4) + ml;

    v8f c = {};
    for (int k0 = 0; k0 < D_HID; k0 += 4) {
        v2f a = *(const v2f*)(H1 + m * D_HID + k0 + kh);
        a.x = fmaxf(a.x, 0.0f);           // ReLU fused into operand load
        a.y = fmaxf(a.y, 0.0f);
        v2f b;
        b.x = W[(size_t)(k0 + kh)     * D_OUT2 + n];
        b.y = W[(size_t)(k0 + kh + 1) * D_OUT2 + n];
        c = __builtin_amdgcn_wmma_f32_16x16x4_f32(false, a, false, b,
                                                  (short)0, c, false, false);
    }
    const int rbase = (tileM << 4) + (half << 3);
#pragma unroll
    for (int v = 0; v < 8; ++v)
        Z0[(size_t)(rbase + v) * D_OUT2 + n] = c[v];
}

// ---------------------------------------------------------------------------
// SpMM 2: z[row] += val * z0[col], 64 channels. 4 edges per 256-thr block.
// ---------------------------------------------------------------------------
__global__ void spmm64_kernel(const int*   __restrict__ erow,
                              const int*   __restrict__ ecol,
                              const float* __restrict__ eval,
                              const float* __restrict__ Z0,
                              float*       __restrict__ Z) {
    const int e = blockIdx.x * 4 + (threadIdx.x >> 6);
    const int c = threadIdx.x & 63;
    const int r = erow[e];
    const int s = ecol[e];
    const float v = eval[e];
    atomicAdd(&Z[(size_t)r * D_OUT2 + c], v * Z0[(size_t)s * D_OUT2 + c]);
}

// ---------------------------------------------------------------------------
// GEMM 3: adj[16384,16384] = z @ z.T.   B(k,n) = z[n][k] -> both operands are
// contiguous float2 loads from z. Output is a 1.07 GB stream -> NT stores.
// ---------------------------------------------------------------------------
__global__ void gemm_zzT_kernel(const float* __restrict__ Z,
                                float* __restrict__ ADJ) {
    const int wave  = (int)((blockIdx.x * blockDim.x + threadIdx.x) >> 5);
    const int lane  = threadIdx.x & 31;
    const int tileM = wave >> 10;         // 16384/16 = 1024 N-tiles
    const int tileN = wave & 1023;
    const int ml    = lane & 15;
    const int half  = lane >> 4;
    const int kh    = half << 1;
    const size_t ma = (size_t)(tileM << 4) + ml;   // A row in z
    const size_t nb = (size_t)(tileN << 4) + ml;   // B col = z row

    v8f c = {};
#pragma unroll
    for (int k0 = 0; k0 < D_OUT2; k0 += 4) {       // 16 WMMA ops
        v2f a = *(const v2f*)(Z + ma * D_OUT2 + k0 + kh);
        v2f b = *(const v2f*)(Z + nb * D_OUT2 + k0 + kh);
        c = __builtin_amdgcn_wmma_f32_16x16x4_f32(false, a, false, b,
                                                  (short)0, c, false, false);
    }
    const size_t rbase = (size_t)(tileM << 4) + (half << 3);
    const size_t col   = (size_t)(tileN << 4) + ml;
#pragma unroll
    for (int v = 0; v < 8; ++v)
        __builtin_nontemporal_store(c[v], &ADJ[(rbase + v) * (size_t)N_NODES + col]);
}

// ---------------------------------------------------------------------------
// Launch
// ---------------------------------------------------------------------------
extern "C" void kernel_launch(void* const* d_in, const int* in_sizes, int n_in,
                              void* d_out, int out_size, void* d_ws, size_t ws_size,
                              hipStream_t stream) {
    const float* x    = (const float*)d_in[0];   // [16384,512]
    const int*   erow = (const int*)  d_in[1];   // [E]
    const int*   ecol = (const int*)  d_in[2];   // [E]
    const float* eval = (const float*)d_in[3];   // [E]
    const float* W1   = (const float*)d_in[4];   // [512,256]
    const float* W2   = (const float*)d_in[5];   // [256,64]

    float* out = (float*)d_out;
    float* adj = out;                                     // [16384,16384]
    float* z   = out + (size_t)N_NODES * N_NODES;         // [16384,64]

    float* ws = (float*)d_ws;
    float* h0 = ws;                                       // [16384,256]
    float* h1 = ws + (size_t)N_NODES * D_HID;             // [16384,256]
    float* z0 = ws + 2 * (size_t)N_NODES * D_HID;         // [16384,64]

    // 1) h0 = x @ W1           (16384 tiles, 8 waves/block)
    gemm_xW1_kernel<<<(N_NODES / 16) * (D_HID / 16) / 8, 256, 0, stream>>>(x, W1, h0);

    // 2) h1 = 0 ; scatter-add
    {
        size_t n = (size_t)N_NODES * D_HID;
        fill_zero_kernel<<<(unsigned)((n + 255) / 256), 256, 0, stream>>>(h1, n);
    }
    spmm256_kernel<<<N_EDGES, D_HID, 0, stream>>>(erow, ecol, eval, h0, h1);

    // 3) z0 = relu(h1) @ W2    (4096 tiles)
    gemm_hW2_kernel<<<(N_NODES / 16) * (D_OUT2 / 16) / 8, 256, 0, stream>>>(h1, W2, z0);

    // 4) z = 0 ; scatter-add   (z lives in d_out, second tuple element)
    {
        size_t n = (size_t)N_NODES * D_OUT2;
        fill_zero_kernel<<<(unsigned)((n + 255) / 256), 256, 0, stream>>>(z, n);
    }
    spmm64_kernel<<<N_EDGES / 4, 256, 0, stream>>>(erow, ecol, eval, z0, z);

    // 5) adj = z @ z.T         (1,048,576 tiles, 8 waves/block)
    gemm_zzT_kernel<<<(N_NODES / 16) * (N_NODES / 16) / 8, 256, 0, stream>>>(z, adj);
}